// GCNLayer_66262755443071
// MI455X (gfx1250) — compile-verified
//
#include <hip/hip_runtime.h>

typedef __attribute__((ext_vector_type(2))) float v2f;
typedef __attribute__((ext_vector_type(8))) float v8f;

#define DFEAT 64
#define XSTR 68   // LDS row stride for x tile (bank-conflict-free for A-frag reads)
#define WSTR 66   // LDS row stride for W (bank-conflict-free for B-frag reads)

// ---------------- utility: vectorized zero ----------------
__global__ void zero_f128(float4* __restrict__ p, int n4) {
    int i = blockIdx.x * blockDim.x + threadIdx.x;
    if (i < n4) p[i] = make_float4(0.f, 0.f, 0.f, 0.f);
}

// ---------------- degree accumulation (L2 fp32 atomics) ----------------
__global__ void degree_kernel(const long long* __restrict__ ei,
                              const float* __restrict__ ew,
                              float* __restrict__ deg, int E) {
    int e = blockIdx.x * blockDim.x + threadIdx.x;
    if (e < E) {
        int d = (int)ei[(long long)E + e];   // dst row of edge_index
        atomicAdd(&deg[d], ew[e]);
    }
}

// dinv = rsqrt(deg + 1)  (self-loop weight 1 folded in; deg+1 > 0 always)
__global__ void dinv_kernel(float* __restrict__ deg, int n) {
    int i = blockIdx.x * blockDim.x + threadIdx.x;
    if (i < n) deg[i] = rsqrtf(deg[i] + 1.0f);
}

// ---------------- h = x @ W^T via V_WMMA_F32_16X16X4_F32 ----------------
// 8 waves per block; each wave computes a 16-row x 64-col tile of h.
__global__ __launch_bounds__(256)
void gemm_wmma(const float* __restrict__ x, const float* __restrict__ Wg,
               float* __restrict__ h, int nrows) {
    __shared__ float sW[64 * WSTR];          // 16.5 KB
    __shared__ float sX[8 * 16 * XSTR];      // 34 KB

    const int tid  = threadIdx.x;
    const int wave = tid >> 5;
    const int lane = tid & 31;

    // Stage W (64x64) into LDS, padded stride
    #pragma unroll
    for (int i = tid; i < 64 * 64; i += 256)
        sW[(i >> 6) * WSTR + (i & 63)] = Wg[i];

    const int tiles = nrows >> 4;            // N is a multiple of 16
    const int tile  = blockIdx.x * 8 + wave;
    float* sXw = sX + wave * 16 * XSTR;

    if (tile < tiles) {
        // Stage this wave's 16x64 x-tile, coalesced b128 loads
        const float4* xg = (const float4*)(x + (size_t)tile * 16 * DFEAT);
        #pragma unroll
        for (int j = 0; j < 8; ++j) {
            int i4 = lane + j * 32;          // 0..255 float4s
            int r = i4 >> 4, c4 = i4 & 15;
            *(float4*)(sXw + r * XSTR + c4 * 4) = xg[i4];
        }
    }
    __syncthreads();
    if (tile >= tiles) return;

    v8f acc[4] = {};                          // 4 N-tiles of 16 cols
    const int mr  = lane & 15;
    const int khi = (lane >> 4) << 1;         // lanes 16-31 hold K+2,K+3

    #pragma unroll
    for (int kt = 0; kt < 16; ++kt) {
        const int kk = kt * 4 + khi;
        // A-frag: x[m0+mr, kk..kk+1]  (ISA 16x4 f32 A layout)
        v2f a = *(const v2f*)(sXw + mr * XSTR + kk);
        #pragma unroll
        for (int j = 0; j < 4; ++j) {
            // B-frag: B[k,n] = W[n,k] -> W[j*16+mr, kk..kk+1]
            v2f b = *(const v2f*)(sW + (j * 16 + mr) * WSTR + kk);
            acc[j] = __builtin_amdgcn_wmma_f32_16x16x4_f32(
                false, a, false, b, (short)0, acc[j], false, false);
        }
    }

    // Store per documented 16x16 f32 C/D layout:
    // VGPR r: lanes 0-15 -> M=r, lanes 16-31 -> M=r+8; N = lane&15
    const int m0 = tile * 16;
    const int rowoff = (lane >> 4) << 3;
    const int col = lane & 15;
    #pragma unroll
    for (int r = 0; r < 8; ++r) {
        int row = m0 + r + rowoff;
        #pragma unroll
        for (int j = 0; j < 4; ++j)
            h[(size_t)row * DFEAT + j * 16 + col] = acc[j][r];
    }
}

// ---------------- edge scatter: one wave32 per edge ----------------
__global__ __launch_bounds__(256)
void scatter_kernel(const long long* __restrict__ ei,
                    const float* __restrict__ ew,
                    const float* __restrict__ dinv,
                    const float* __restrict__ h,
                    float* __restrict__ out, int E) {
    long long gid = (long long)blockIdx.x * blockDim.x + threadIdx.x;
    long long e = gid >> 5;
    int lane = (int)(gid & 31);
    if (e >= E) return;
    int s = (int)ei[e];
    int d = (int)ei[(long long)E + e];
    float norm = dinv[s] * ew[e] * dinv[d];
    const float2 hv = *(const float2*)(h + (size_t)s * DFEAT + lane * 2);
    float* o = out + (size_t)d * DFEAT + lane * 2;
    atomicAdd(o + 0, hv.x * norm);
    atomicAdd(o + 1, hv.y * norm);
}

// ---------------- fused self-loop + bias + ReLU ----------------
__global__ void finalize_kernel(const float* __restrict__ h,
                                const float* __restrict__ dinv,
                                const float* __restrict__ b,
                                float* __restrict__ out, int n) {
    int idx = blockIdx.x * blockDim.x + threadIdx.x;  // one float2 per thread
    if (idx >= n * 32) return;
    int i = idx >> 5;
    int c = (idx & 31) * 2;
    float s = dinv[i]; s = s * s;                     // self-loop norm
    float2 o  = *(float2*)(out + (size_t)i * DFEAT + c);
    float2 hv = *(const float2*)(h + (size_t)i * DFEAT + c);
    o.x = fmaxf(o.x + hv.x * s + b[c], 0.f);
    o.y = fmaxf(o.y + hv.y * s + b[c + 1], 0.f);
    *(float2*)(out + (size_t)i * DFEAT + c) = o;
}

extern "C" void kernel_launch(void* const* d_in, const int* in_sizes, int n_in,
                              void* d_out, int out_size, void* d_ws, size_t ws_size,
                              hipStream_t stream) {
    const float*     x  = (const float*)d_in[0];
    const long long* ei = (const long long*)d_in[1];
    const float*     ew = (const float*)d_in[2];
    const float*     W  = (const float*)d_in[3];
    const float*     b  = (const float*)d_in[4];
    float* out = (float*)d_out;

    const int N = in_sizes[0] / DFEAT;
    const int E = in_sizes[1] / 2;

    // workspace layout: [deg/dinv: N floats][h: N*64 floats]
    float* deg = (float*)d_ws;
    float* h   = deg + ((N + 127) & ~127);

    // zero out accumulator + degree (determinism across graph replays)
    int n4o = (N * DFEAT) / 4;
    zero_f128<<<(n4o + 255) / 256, 256, 0, stream>>>((float4*)out, n4o);
    int n4d = (N + 3) / 4;
    zero_f128<<<(n4d + 255) / 256, 256, 0, stream>>>((float4*)deg, n4d);

    degree_kernel<<<(E + 255) / 256, 256, 0, stream>>>(ei, ew, deg, E);
    dinv_kernel<<<(N + 255) / 256, 256, 0, stream>>>(deg, N);

    int tiles = N / 16;
    gemm_wmma<<<(tiles + 7) / 8, 256, 0, stream>>>(x, W, h, N);

    long long sthreads = (long long)E * 32;
    scatter_kernel<<<(unsigned)((sthreads + 255) / 256), 256, 0, stream>>>(
        ei, ew, deg, h, out, E);

    finalize_kernel<<<(N * 32 + 255) / 256, 256, 0, stream>>>(h, deg, b, out, N);
}